// EmbeddingP_39479339385295
// MI455X (gfx1250) — compile-verified
//
#include <hip/hip_runtime.h>
#include <hip/hip_bf16.h>

#define N_NODES   100000
#define N_EDGES   1600000
#define FEAT_DIM  128
#define EMBED_DIM 32
#define NCLS      17          // N_CLASS_P1

typedef __attribute__((ext_vector_type(2))) float v2f;
typedef __attribute__((ext_vector_type(8))) float v8f;

// ---------------------------------------------------------------------------
// K0: zero the recall_node accumulation region (atomics add into it).
// ---------------------------------------------------------------------------
__global__ void zero_kernel(float* __restrict__ p, int n) {
    int i = blockIdx.x * blockDim.x + threadIdx.x;
    if (i < n) p[i] = 0.0f;
}

// ---------------------------------------------------------------------------
// K1: embed = features @ W_embed + b_embed   (100000x128 @ 128x32)
// One wave per 16-row tile; two 16-wide N tiles; K=128 -> 32 f32 WMMAs.
// A 16x4 layout: lane m=(lane&15) holds row m; koff=(lane>>4)*2 -> K pair.
// B 4x16 layout mirrors A with n=(lane&15).
// C/D: VGPR r, lanes 0-15 = M=r, lanes 16-31 = M=r+8, N=lane&15.
// ---------------------------------------------------------------------------
__global__ void embed_gemm(const float* __restrict__ F,
                           const float* __restrict__ W,
                           const float* __restrict__ bias,
                           float* __restrict__ E) {
    const int lane = threadIdx.x;            // 0..31 (wave32)
    const int row0 = blockIdx.x * 16;
    const int m    = lane & 15;
    const int koff = (lane >> 4) * 2;        // 0 or 2
    const int n    = lane & 15;

    const float* __restrict__ frow = F + (size_t)(row0 + m) * FEAT_DIM;

    v8f acc0 = {};
    v8f acc1 = {};
    for (int k = 0; k < FEAT_DIM; k += 4) {
        v2f a;
        a.x = frow[k + koff];
        a.y = frow[k + koff + 1];
        v2f b0, b1;
        b0.x = W[(k + koff)     * EMBED_DIM + n];
        b0.y = W[(k + koff + 1) * EMBED_DIM + n];
        b1.x = W[(k + koff)     * EMBED_DIM + 16 + n];
        b1.y = W[(k + koff + 1) * EMBED_DIM + 16 + n];
        acc0 = __builtin_amdgcn_wmma_f32_16x16x4_f32(false, a, false, b0,
                                                     (short)0, acc0, false, false);
        acc1 = __builtin_amdgcn_wmma_f32_16x16x4_f32(false, a, false, b1,
                                                     (short)0, acc1, false, false);
    }

    const int hi = lane >> 4;                // 0 or 1
    #pragma unroll
    for (int r = 0; r < 8; ++r) {
        int row = row0 + r + hi * 8;
        E[(size_t)row * EMBED_DIM + n]      = acc0[r] + bias[n];
        E[(size_t)row * EMBED_DIM + 16 + n] = acc1[r] + bias[16 + n];
    }
}

// ---------------------------------------------------------------------------
// K2: per-edge transform. 8 waves/block, each wave owns 16 edges.
//  - gather src/dst embeddings (coalesced 128B rows), build E_feat in LDS
//  - E_feat(16x64) @ W_trans(64x17, zero-padded to 32 cols) via f32 WMMA
//  - softmax over 17 classes, write poss_edge, atomicAdd into recall[src]
// ---------------------------------------------------------------------------
#define WAVES_PER_BLOCK 8
#define EF_PAD 68   // row stride (floats); 68%64=4 -> conflict-free A reads

__global__ void edge_kernel(const float* __restrict__ embed,
                            const int*   __restrict__ edges,
                            const float* __restrict__ weights,
                            const float* __restrict__ Wt,
                            const float* __restrict__ bt,
                            float* __restrict__ poss_edge,
                            float* __restrict__ recall) {
    __shared__ float Bs[64][32];                       // padded W_trans
    __shared__ float Ef[WAVES_PER_BLOCK][16][EF_PAD];  // E_feat tiles
    __shared__ float Lg[WAVES_PER_BLOCK][16][20];      // logits (17 + pad)

    const int tid  = threadIdx.x;
    const int wid  = tid >> 5;
    const int lane = tid & 31;

    // cooperative load of zero-padded W_trans (64x17 -> 64x32)
    for (int i = tid; i < 64 * 32; i += blockDim.x) {
        int kk = i >> 5, nn = i & 31;
        Bs[kk][nn] = (nn < NCLS) ? Wt[kk * NCLS + nn] : 0.0f;
    }

    const long base = ((long)blockIdx.x * WAVES_PER_BLOCK + wid) * 16;

    // build E_feat: lane covers one of 32 embed components per edge
    for (int e = 0; e < 16; ++e) {
        long ei = base + e;
        int s = edges[2 * ei];
        int d = edges[2 * ei + 1];
        float se = embed[(size_t)s * EMBED_DIM + lane];
        float de = embed[(size_t)d * EMBED_DIM + lane];
        Ef[wid][e][lane]      = 0.5f * (se + de);
        float df = se - de;
        Ef[wid][e][32 + lane] = df * df;
    }
    __syncthreads();

    // WMMA: 16 edges x 17 classes, K=64 -> 16 steps, 2 N tiles
    const int m    = lane & 15;
    const int koff = (lane >> 4) * 2;
    const int n    = lane & 15;
    v8f c0 = {};
    v8f c1 = {};
    for (int k = 0; k < 64; k += 4) {
        v2f a;
        a.x = Ef[wid][m][k + koff];
        a.y = Ef[wid][m][k + koff + 1];
        v2f b0, b1;
        b0.x = Bs[k + koff][n];
        b0.y = Bs[k + koff + 1][n];
        b1.x = Bs[k + koff][16 + n];
        b1.y = Bs[k + koff + 1][16 + n];
        c0 = __builtin_amdgcn_wmma_f32_16x16x4_f32(false, a, false, b0,
                                                   (short)0, c0, false, false);
        c1 = __builtin_amdgcn_wmma_f32_16x16x4_f32(false, a, false, b1,
                                                   (short)0, c1, false, false);
    }

    const int hi = lane >> 4;
    #pragma unroll
    for (int r = 0; r < 8; ++r) {
        int row = r + hi * 8;
        Lg[wid][row][n] = c0[r] + bt[n];                 // classes 0..15
        if (n == 0)
            Lg[wid][row][16] = c1[r] + bt[16];           // class 16
    }
    __syncthreads();

    // softmax + outputs: lanes 0..15 each own one edge
    if (lane < 16) {
        long ei = base + lane;
        float mx = -3.4e38f;
        #pragma unroll
        for (int c = 0; c < NCLS; ++c) mx = fmaxf(mx, Lg[wid][lane][c]);
        float ex[NCLS];
        float sum = 0.0f;
        #pragma unroll
        for (int c = 0; c < NCLS; ++c) {
            ex[c] = __expf(Lg[wid][lane][c] - mx);
            sum += ex[c];
        }
        float inv = 1.0f / sum;
        float w   = weights[ei];
        int   s   = edges[2 * ei];
        float* __restrict__ pe = poss_edge + (size_t)ei * NCLS;
        float* __restrict__ rc = recall    + (size_t)s  * NCLS;
        #pragma unroll
        for (int c = 0; c < NCLS; ++c) {
            float p = ex[c] * inv;
            pe[c] = p;
            atomicAdd(&rc[c], p * w);
        }
    }
}

// ---------------------------------------------------------------------------
// K3: poss_node = recall_node / neighbours_sum
// ---------------------------------------------------------------------------
__global__ void node_kernel(const float* __restrict__ recall,
                            const float* __restrict__ nsum,
                            float* __restrict__ poss_node, int n) {
    int i = blockIdx.x * blockDim.x + threadIdx.x;
    if (i < n) {
        int node = i / NCLS;
        poss_node[i] = recall[i] / nsum[node];
    }
}

// ---------------------------------------------------------------------------
extern "C" void kernel_launch(void* const* d_in, const int* in_sizes, int n_in,
                              void* d_out, int out_size, void* d_ws, size_t ws_size,
                              hipStream_t stream) {
    const float* features = (const float*)d_in[0];
    const int*   edges    = (const int*)  d_in[1];
    const float* weights  = (const float*)d_in[2];
    const float* nsum     = (const float*)d_in[3];
    const float* W_embed  = (const float*)d_in[4];
    const float* b_embed  = (const float*)d_in[5];
    const float* W_trans  = (const float*)d_in[6];
    const float* b_trans  = (const float*)d_in[7];

    float* out       = (float*)d_out;
    float* poss_node = out;                                        // N_NODES*17
    float* poss_edge = out + (size_t)N_NODES * NCLS;               // N_EDGES*17
    float* recall    = out + (size_t)N_NODES * NCLS
                           + (size_t)N_EDGES * NCLS;               // N_NODES*17
    float* embed     = (float*)d_ws;                               // N_NODES*32 f32

    const int node_elems = N_NODES * NCLS;                         // 1,700,000

    zero_kernel<<<(node_elems + 255) / 256, 256, 0, stream>>>(recall, node_elems);

    embed_gemm<<<N_NODES / 16, 32, 0, stream>>>(features, W_embed, b_embed, embed);

    edge_kernel<<<N_EDGES / (16 * WAVES_PER_BLOCK), 32 * WAVES_PER_BLOCK, 0, stream>>>(
        embed, edges, weights, W_trans, b_trans, poss_edge, recall);

    node_kernel<<<(node_elems + 255) / 256, 256, 0, stream>>>(recall, nsum,
                                                              poss_node, node_elems);
}